// FAGCN_33603824124470
// MI455X (gfx1250) — compile-verified
//
#include <hip/hip_runtime.h>
#include <hip/hip_bf16.h>
#include <math.h>

typedef __attribute__((ext_vector_type(2))) float v2f;
typedef __attribute__((ext_vector_type(8))) float v8f;

#define EPS_F 0.3f
#define HID 128
#define OUTC 64
#define INC 256

// ---------------------------------------------------------------------------
// Generic zero-fill
// ---------------------------------------------------------------------------
__global__ void zero_kernel(float* __restrict__ p, size_t n) {
    size_t i = (size_t)blockIdx.x * blockDim.x + threadIdx.x;
    size_t stride = (size_t)gridDim.x * blockDim.x;
    for (; i < n; i += stride) p[i] = 0.0f;
}

// ---------------------------------------------------------------------------
// WMMA f32 GEMM:  Out[M,N] = relu?( A[M,K] @ W[N,K]^T + bias[N] )
// One wave computes one 16x16 tile using V_WMMA_F32_16X16X4_F32.
// A fragment (16x4, 32-bit): lane<16: M=lane, K={k0,k0+1}; lane>=16: M=lane-16,
// K={k0+2,k0+3}. B fragment symmetric (rows of B striped across lanes).
// C/D: VGPR r, lane L -> M = r + 8*(L>>4), N = L&15.
// blockDim.x = 128 (4 waves); wave w covers n-tile (blockIdx.y*4 + w).
// ---------------------------------------------------------------------------
__global__ void wmma_gemm_kernel(const float* __restrict__ A,
                                 const float* __restrict__ W,
                                 const float* __restrict__ bias,
                                 float* __restrict__ Out,
                                 int M, int K, int N, int do_relu) {
    const int wave = threadIdx.x >> 5;
    const int lane = threadIdx.x & 31;
    const int half = lane >> 4;
    const int l16  = lane & 15;

    const int m0 = blockIdx.x * 16;
    const int n0 = (blockIdx.y * 4 + wave) * 16;
    if (n0 >= N) return;                       // uniform per wave

    const float* __restrict__ arow = A + (size_t)(m0 + l16) * K;
    const float* __restrict__ wrow = W + (size_t)(n0 + l16) * K;

    v8f acc = {};
    for (int k0 = 0; k0 < K; k0 += 4) {
        const int ka = k0 + 2 * half;          // even -> 8B aligned
        const float2 af = *(const float2*)(arow + ka);
        const float2 bf = *(const float2*)(wrow + ka);
        v2f a; a.x = af.x; a.y = af.y;
        v2f b; b.x = bf.x; b.y = bf.y;
        acc = __builtin_amdgcn_wmma_f32_16x16x4_f32(
            /*neg_a=*/false, a, /*neg_b=*/false, b,
            /*c_mod=*/(short)0, acc, /*reuse_a=*/false, /*reuse_b=*/false);
    }

    const float bv = bias[n0 + l16];
#pragma unroll
    for (int r = 0; r < 8; ++r) {
        float v = acc[r] + bv;
        if (do_relu) v = fmaxf(v, 0.0f);
        Out[(size_t)(m0 + r + 8 * half) * N + (n0 + l16)] = v;
    }
}

// ---------------------------------------------------------------------------
// Degree count: deg[row[e]] += 1
// ---------------------------------------------------------------------------
__global__ void deg_kernel(const long long* __restrict__ row,
                           float* __restrict__ deg, long long n_edges) {
    long long e = (long long)blockIdx.x * blockDim.x + threadIdx.x;
    if (e >= n_edges) return;
    atomicAdd(&deg[(int)row[e]], 1.0f);
}

// deg -> dinv (in place): dinv = deg>0 ? rsqrt(deg) : 0
__global__ void dinv_kernel(float* __restrict__ deg, int n) {
    int i = blockIdx.x * blockDim.x + threadIdx.x;
    if (i >= n) return;
    float d = deg[i];
    deg[i] = (d > 0.0f) ? rsqrtf(fmaxf(d, 1.0f)) : 0.0f;
}

// ---------------------------------------------------------------------------
// Per-node gate projections: si[n] = h[n,:]·gw_i, sj[n] = h[n,:]·gw_j
// One wave per node, 4 floats (one float4) per lane, shfl-reduce.
// ---------------------------------------------------------------------------
__global__ void sisj_kernel(const float* __restrict__ h,
                            const float* __restrict__ gate_w_l,   // 256: [gw_i|gw_j]
                            float* __restrict__ si, float* __restrict__ sj,
                            int n_nodes) {
    int node = (int)(((size_t)blockIdx.x * blockDim.x + threadIdx.x) >> 5);
    int lane = threadIdx.x & 31;
    if (node >= n_nodes) return;

    const float4 hv = *(const float4*)(h + (size_t)node * HID + lane * 4);
    const float4 gi = *(const float4*)(gate_w_l + lane * 4);
    const float4 gj = *(const float4*)(gate_w_l + HID + lane * 4);

    float di = hv.x * gi.x + hv.y * gi.y + hv.z * gi.z + hv.w * gi.w;
    float dj = hv.x * gj.x + hv.y * gj.y + hv.z * gj.z + hv.w * gj.w;
#pragma unroll
    for (int off = 16; off > 0; off >>= 1) {
        di += __shfl_down(di, off, 32);
        dj += __shfl_down(dj, off, 32);
    }
    if (lane == 0) { si[node] = di; sj[node] = dj; }
}

// ---------------------------------------------------------------------------
// Edge scatter: agg[row] += tanh(si[row]+sj[col]+gb[l]) * dinv[row]*dinv[col] * h[col]
// One wave per edge; one float4 of h[col] per lane; 4 f32 atomics per lane.
// ---------------------------------------------------------------------------
__global__ void scatter_kernel(const long long* __restrict__ row,
                               const long long* __restrict__ col,
                               const float* __restrict__ h,
                               const float* __restrict__ si,
                               const float* __restrict__ sj,
                               const float* __restrict__ dinv,
                               const float* __restrict__ gate_b, int layer,
                               float* __restrict__ agg, long long n_edges) {
    long long e = (((long long)blockIdx.x * blockDim.x) + threadIdx.x) >> 5;
    int lane = threadIdx.x & 31;
    if (e >= n_edges) return;

    const int r = (int)row[e];
    const int c = (int)col[e];
    const float g = tanhf(si[r] + sj[c] + gate_b[layer]);
    const float w = g * dinv[r] * dinv[c];

    const float4 hv = *(const float4*)(h + (size_t)c * HID + lane * 4);
    float* __restrict__ ag = agg + (size_t)r * HID + lane * 4;
    atomicAdd(ag + 0, w * hv.x);
    atomicAdd(ag + 1, w * hv.y);
    atomicAdd(ag + 2, w * hv.z);
    atomicAdd(ag + 3, w * hv.w);
}

// ---------------------------------------------------------------------------
// h_new = EPS*raw + agg ; agg = 0 (ready for next layer)
// ---------------------------------------------------------------------------
__global__ void combine_kernel(const float* __restrict__ raw,
                               float* __restrict__ agg,
                               float* __restrict__ hout, size_t n) {
    size_t i = (size_t)blockIdx.x * blockDim.x + threadIdx.x;
    size_t stride = (size_t)gridDim.x * blockDim.x;
    for (; i < n; i += stride) {
        hout[i] = EPS_F * raw[i] + agg[i];
        agg[i] = 0.0f;
    }
}

// ---------------------------------------------------------------------------
// In-place log_softmax over 64 logits per node. One wave per node, 2/lane.
// ---------------------------------------------------------------------------
__global__ void logsoftmax_kernel(float* __restrict__ out, int n_nodes) {
    int node = (int)(((size_t)blockIdx.x * blockDim.x + threadIdx.x) >> 5);
    int lane = threadIdx.x & 31;
    if (node >= n_nodes) return;

    float* p = out + (size_t)node * OUTC + lane * 2;
    float2 v = *(float2*)p;

    float m = fmaxf(v.x, v.y);
#pragma unroll
    for (int off = 16; off > 0; off >>= 1) m = fmaxf(m, __shfl_xor(m, off, 32));
    float s = expf(v.x - m) + expf(v.y - m);
#pragma unroll
    for (int off = 16; off > 0; off >>= 1) s += __shfl_xor(s, off, 32);
    const float lg = m + logf(s);
    v.x -= lg; v.y -= lg;
    *(float2*)p = v;
}

// ---------------------------------------------------------------------------
extern "C" void kernel_launch(void* const* d_in, const int* in_sizes, int n_in,
                              void* d_out, int out_size, void* d_ws, size_t ws_size,
                              hipStream_t stream) {
    const float*     x   = (const float*)d_in[0];
    const long long* ei  = (const long long*)d_in[1];   // int64 per reference
    const float*     t1w = (const float*)d_in[2];
    const float*     t1b = (const float*)d_in[3];
    const float*     gw  = (const float*)d_in[4];       // [2][256]
    const float*     gb  = (const float*)d_in[5];       // [2]
    const float*     t2w = (const float*)d_in[6];
    const float*     t2b = (const float*)d_in[7];
    float* out = (float*)d_out;

    const int       n  = in_sizes[0] / INC;             // 50000
    const long long ne = (long long)in_sizes[1] / 2;    // 1.6M

    // workspace layout (floats)
    float* ws   = (float*)d_ws;
    const size_t hsz = (size_t)n * HID;
    float* raw  = ws;                 // n*128
    float* hcur = ws + hsz;           // n*128
    float* agg  = ws + 2 * hsz;       // n*128
    float* dinv = ws + 3 * hsz;       // n   (deg then dinv, in place)
    float* si   = dinv + n;           // n
    float* sj   = si + n;             // n

    // zero agg + deg
    zero_kernel<<<2048, 256, 0, stream>>>(agg, hsz);
    zero_kernel<<<(n + 255) / 256, 256, 0, stream>>>(dinv, (size_t)n);

    // h0 = relu(x @ t1_w.T + t1_b)  -> raw   (M=50000 = 3125*16 exactly)
    {
        dim3 grid((n + 15) / 16, HID / 64);
        wmma_gemm_kernel<<<grid, 128, 0, stream>>>(x, t1w, t1b, raw, n, INC, HID, 1);
    }

    // degree -> dinv
    deg_kernel<<<(unsigned)((ne + 255) / 256), 256, 0, stream>>>(ei, dinv, ne);
    dinv_kernel<<<(n + 255) / 256, 256, 0, stream>>>(dinv, n);

    // two FAGCN layers
    for (int l = 0; l < 2; ++l) {
        const float* hsrc = (l == 0) ? raw : hcur;
        sisj_kernel<<<(unsigned)(((size_t)n * 32 + 255) / 256), 256, 0, stream>>>(
            hsrc, gw + l * 2 * HID, si, sj, n);
        scatter_kernel<<<(unsigned)((ne * 32 + 255) / 256), 256, 0, stream>>>(
            ei, ei + ne, hsrc, si, sj, dinv, gb, l, agg, ne);
        combine_kernel<<<2048, 256, 0, stream>>>(raw, agg, hcur, hsz);
    }

    // logits = hcur @ t2_w.T + t2_b  -> out, then in-place log_softmax
    {
        dim3 grid((n + 15) / 16, OUTC / 64);
        wmma_gemm_kernel<<<grid, 128, 0, stream>>>(hcur, t2w, t2b, out, n, HID, OUTC, 0);
    }
    logsoftmax_kernel<<<(unsigned)(((size_t)n * 32 + 255) / 256), 256, 0, stream>>>(out, n);
}